// ProposalGenerator_2869038154305
// MI455X (gfx1250) — compile-verified
//
#include <hip/hip_runtime.h>
#include <hip/hip_bf16.h>
#include <stdint.h>

#define BDIM 256

constexpr int   B_       = 4;
constexpr int   G_       = 25;
constexpr int   A_       = 128 * 128 * 9;   // 147456 anchors
constexpr int   NBLK_    = A_ / BDIM;       // 576 tiles per batch
constexpr int   TILES_   = 4;               // tiles per workgroup (pipelined)
constexpr int   GRIDX_   = NBLK_ / TILES_;  // 144 workgroups per batch
constexpr int   TOT_     = A_;              // 147456 output rows per batch
constexpr int   POS_CAP_ = 16384;           // realistic cap on #positives
constexpr float FM_      = 128.0f;
constexpr float SCALE_   = 0.125f;          // FM_W / TM_W
constexpr float POS_TH_  = 0.7f;
constexpr float NEG_TH_  = 0.3f;
constexpr float MIN_POS_ = 0.15f;           // 0.5 * NEG_TH

typedef __attribute__((ext_vector_type(4))) unsigned int u32x4;
typedef __attribute__((ext_vector_type(8))) int          i32x8;
typedef __attribute__((ext_vector_type(4))) int          i32x4;

#if defined(__HIP_DEVICE_COMPILE__) && \
    __has_builtin(__builtin_amdgcn_tensor_load_to_lds) && \
    __has_builtin(__builtin_amdgcn_s_wait_tensorcnt)
#define HAVE_TDM 1
#else
#define HAVE_TDM 0
#endif

#if HAVE_TDM
// Issue one TDM descriptor: 1024 contiguous f32 (one 4KB anchor tile) -> LDS.
__device__ inline void tdm_issue(const float* gsrc, float* ldst)
{
    unsigned long long ga = (unsigned long long)(uintptr_t)gsrc;
    unsigned lds = (unsigned)(uintptr_t)ldst;
    const int NE = BDIM * 4;                 // 1024 elements
    // D# group0: count=1, lds_addr, 57-bit global addr, type=2
    u32x4 g0 = { 1u, lds, (unsigned)ga,
                 (unsigned)(((ga >> 32) & 0x1FFFFFFull) | 0x80000000ull) };
    // D# group1: data_size=4B; tensor_dim0=tile_dim0=1024; dim1=1
    i32x8 g1 = { (int)(2u << 16),            // data_size code 2 (4 bytes)
                 (int)(NE << 16),            // tensor_dim0[15:0] @ bits 63:48
                 (int)(1  << 16),            // tensor_dim1[15:0] @ bits 95:80
                 (int)(NE << 16),            // tile_dim0         @ bits 127:112
                 1,                          // tile_dim1 = 1
                 NE,                         // tensor_dim0_stride low32
                 (int)(NE << 16),            // tensor_dim1_stride[15:0]
                 0 };
    i32x4 gz = { 0, 0, 0, 0 };
#if __clang_major__ >= 23
    i32x8 g4 = { 0, 0, 0, 0, 0, 0, 0, 0 };
    __builtin_amdgcn_tensor_load_to_lds(g0, g1, gz, gz, g4, 0);
#else
    __builtin_amdgcn_tensor_load_to_lds(g0, g1, gz, gz, 0);
#endif
}
#endif

// ---------------------------------------------------------------------------
// K1: per-(anchor,gt) IoU + pos/neg classification.
//     4 anchor tiles per block, double-buffered TDM pipeline:
//       issue(i+1) -> s_wait_tensorcnt(1) -> compute(i)
// ---------------------------------------------------------------------------
__global__ void iou_mask_kernel(const float* __restrict__ anchors,
                                const float* __restrict__ gt,
                                unsigned* __restrict__ negcnt,
                                unsigned* __restrict__ negmask,
                                float* __restrict__ posiou,
                                unsigned* __restrict__ posidx,
                                unsigned* __restrict__ counters)
{
    const int b = blockIdx.y, blk = blockIdx.x, t = threadIdx.x;
    __shared__ float sAnc[2][BDIM * 4];
    __shared__ float sGT[G_ * 4];

    const float* base =
        anchors + ((size_t)b * A_ + (size_t)blk * TILES_ * BDIM) * 4;

    if (t < G_ * 4) sGT[t] = gt[(size_t)b * G_ * 4 + t] * SCALE_;
    if (t == 0 && blk + 1 < gridDim.x)
        __builtin_prefetch(base + (size_t)TILES_ * BDIM * 4, 0, 1);

#if HAVE_TDM
    if (t < 32) tdm_issue(base, &sAnc[0][0]);      // wave 0 only, wave-uniform
#else
    ((float4*)sAnc[0])[t] = ((const float4*)base)[t];
#endif

#pragma unroll
    for (int i = 0; i < TILES_; ++i) {
#if HAVE_TDM
        if (t < 32) {
            if (i + 1 < TILES_) {
                tdm_issue(base + (size_t)(i + 1) * BDIM * 4,
                          &sAnc[(i + 1) & 1][0]);
                __builtin_amdgcn_s_wait_tensorcnt(1);  // tile i done, i+1 in flight
            } else {
                __builtin_amdgcn_s_wait_tensorcnt(0);
            }
        }
#endif
        __syncthreads();
        const float* sA = &sAnc[i & 1][0];

        const float ax1 = sA[t * 4 + 0], ay1 = sA[t * 4 + 1];
        const float ax2 = sA[t * 4 + 2], ay2 = sA[t * 4 + 3];
        const bool valid =
            (ax1 >= 0.f) && (ay1 >= 0.f) && (ax2 <= FM_) && (ay2 <= FM_);
        const float areaA = (ax2 - ax1) * (ay2 - ay1);

        float iou[G_];
        float rmax = -1.0f;
#pragma unroll
        for (int g = 0; g < G_; ++g) {
            float gx1 = sGT[g * 4 + 0], gy1 = sGT[g * 4 + 1];
            float gx2 = sGT[g * 4 + 2], gy2 = sGT[g * 4 + 3];
            float lx = fmaxf(ax1, gx1), ly = fmaxf(ay1, gy1);
            float rx = fminf(ax2, gx2), ry = fminf(ay2, gy2);
            float inter = fmaxf(rx - lx, 0.f) * fmaxf(ry - ly, 0.f);
            float areaB = (gx2 - gx1) * (gy2 - gy1);
            float v = inter / (areaA + areaB - inter);
            iou[g] = v;
            rmax = fmaxf(rmax, v);
        }

        const int a = (blk * TILES_ + i) * BDIM + t;
        unsigned nm = 0, nc = 0;
#pragma unroll
        for (int g = 0; g < G_; ++g) {
            bool pos = valid && (((iou[g] == rmax) && (iou[g] > MIN_POS_)) ||
                                 (iou[g] >= POS_TH_));
            bool neg = valid && (iou[g] < NEG_TH_) && !pos;
            if (pos) {
                unsigned slot = atomicAdd(&counters[b * 8 + 0], 1u);
                if (slot < (unsigned)POS_CAP_) {
                    posiou[(size_t)b * POS_CAP_ + slot] = iou[g];
                    posidx[(size_t)b * POS_CAP_ + slot] =
                        (unsigned)(a * G_ + g);
                }
            }
            if (neg) { nm |= (1u << g); ++nc; }
        }
        negcnt [(size_t)b * A_ + a] = nc;
        negmask[(size_t)b * A_ + a] = nm;
        __syncthreads();   // all reads of sAnc[i&1] done before it is re-filled
#if !HAVE_TDM
        if (i + 1 < TILES_)
            ((float4*)sAnc[(i + 1) & 1])[t] =
                ((const float4*)(base + (size_t)(i + 1) * BDIM * 4))[t];
#endif
    }
}

// ---------------------------------------------------------------------------
// K2/K3: two-level exclusive scan of per-anchor negative counts.
// ---------------------------------------------------------------------------
__global__ void scan_block_kernel(const unsigned* __restrict__ negcnt,
                                  unsigned* __restrict__ negpre,
                                  unsigned* __restrict__ blocksum)
{
    const int b = blockIdx.y, blk = blockIdx.x, t = threadIdx.x;
    const int a = blk * BDIM + t;
    __shared__ unsigned s[BDIM];
    unsigned v = negcnt[(size_t)b * A_ + a];
    s[t] = v; __syncthreads();
    for (int off = 1; off < BDIM; off <<= 1) {
        unsigned u = (t >= off) ? s[t - off] : 0u;
        __syncthreads();
        s[t] += u; __syncthreads();
    }
    negpre[(size_t)b * A_ + a] = s[t] - v;
    if (t == BDIM - 1) blocksum[b * NBLK_ + blk] = s[t];
}

__global__ void scan_top_kernel(unsigned* __restrict__ blocksum,
                                unsigned* __restrict__ counters)
{
    const int b = blockIdx.x, t = threadIdx.x;   // blockDim == NBLK_ (576)
    __shared__ unsigned s[NBLK_];
    unsigned v = blocksum[b * NBLK_ + t];
    s[t] = v; __syncthreads();
    for (int off = 1; off < NBLK_; off <<= 1) {
        unsigned u = (t >= off) ? s[t - off] : 0u;
        __syncthreads();
        s[t] += u; __syncthreads();
    }
    blocksum[b * NBLK_ + t] = s[t] - v;          // global exclusive offset
    if (t == NBLK_ - 1) counters[b * 8 + 2] = s[t];   // total #negatives
}

// ---------------------------------------------------------------------------
// K4: rank-sort positives (IoU desc, flat idx asc) == lax.top_k tie rules.
// ---------------------------------------------------------------------------
__global__ void sort_pos_kernel(const float* __restrict__ posiou,
                                const unsigned* __restrict__ posidx,
                                unsigned* __restrict__ possort,
                                unsigned* __restrict__ counters)
{
    const int b = blockIdx.x, t = threadIdx.x;
    unsigned n = counters[b * 8 + 0];
    if (n > (unsigned)POS_CAP_) n = POS_CAP_;
    for (unsigned i = t; i < n; i += blockDim.x) {
        float ki = posiou[(size_t)b * POS_CAP_ + i];
        unsigned xi = posidx[(size_t)b * POS_CAP_ + i];
        unsigned rank = 0;
        for (unsigned q = 0; q < n; ++q) {
            float kq = posiou[(size_t)b * POS_CAP_ + q];
            unsigned xq = posidx[(size_t)b * POS_CAP_ + q];
            rank += (kq > ki) || ((kq == ki) && (xq < xi));
        }
        possort[(size_t)b * POS_CAP_ + rank] = xi;
    }
    if (t == 0) counters[b * 8 + 1] = n;         // m
}

// ---------------------------------------------------------------------------
// Row emitter: anchors, obj, cls, regression offsets (b128 stores).
// ---------------------------------------------------------------------------
__device__ inline void write_row(float* __restrict__ out,
                                 const float* __restrict__ anchors,
                                 const float* __restrict__ gt,
                                 const int* __restrict__ cls,
                                 int b, long long j, int sel, float obj)
{
    const int ai = sel / G_, gi = sel % G_;
    const float4 av = ((const float4*)anchors)[(size_t)b * A_ + ai];
    const float4 gv = ((const float4*)gt)[(size_t)b * G_ + gi];
    float ax1 = av.x, ay1 = av.y, ax2 = av.z, ay2 = av.w;
    float gx1 = gv.x * SCALE_, gy1 = gv.y * SCALE_;
    float gx2 = gv.z * SCALE_, gy2 = gv.w * SCALE_;
    float acx = (ax1 + ax2) * 0.5f, acy = (ay1 + ay2) * 0.5f;
    float aw = ax2 - ax1, ah = ay2 - ay1;
    float gcx = (gx1 + gx2) * 0.5f, gcy = (gy1 + gy2) * 0.5f;
    float gw = gx2 - gx1, gh = gy2 - gy1;
    float tx = (gcx - acx) / aw, ty = (gcy - acy) / ah;
    float tw = __logf(gw / aw), th = __logf(gh / ah);

    const size_t NT = (size_t)B_ * TOT_;
    const size_t r = (size_t)b * TOT_ + (size_t)j;
    ((float4*)out)[r] = av;                              // sel_anc
    out[NT * 4 + r] = obj;                               // obj
    out[NT * 5 + r] = (float)cls[b * G_ + gi];           // cls
    ((float4*)(out + NT * 6))[r] = make_float4(tx, ty, tw, th);  // offsets
}

__global__ void write_pos_kernel(const float* __restrict__ anchors,
                                 const float* __restrict__ gt,
                                 const int* __restrict__ cls,
                                 const unsigned* __restrict__ possort,
                                 const unsigned* __restrict__ counters,
                                 float* __restrict__ out)
{
    const int b = blockIdx.y;
    const int j = blockIdx.x * BDIM + threadIdx.x;
    const unsigned m = counters[b * 8 + 1];
    if ((unsigned)j < m)
        write_row(out, anchors, gt, cls, b, j,
                  (int)possort[(size_t)b * POS_CAP_ + j], 1.0f);
}

__global__ void write_neg_kernel(const float* __restrict__ anchors,
                                 const float* __restrict__ gt,
                                 const int* __restrict__ cls,
                                 const unsigned* __restrict__ negpre,
                                 const unsigned* __restrict__ blocksum,
                                 const unsigned* __restrict__ negmask,
                                 const unsigned* __restrict__ counters,
                                 float* __restrict__ out)
{
    const int b = blockIdx.y, blk = blockIdx.x, t = threadIdx.x;
    const int a = blk * BDIM + t;
    const unsigned m = counters[b * 8 + 1];
    const unsigned p = negpre[(size_t)b * A_ + a] + blocksum[b * NBLK_ + blk];
    const unsigned mask = negmask[(size_t)b * A_ + a];
    unsigned k = 0;
    for (int g = 0; g < G_; ++g) {
        if (mask & (1u << g)) {
            long long j = (long long)m + p + k;
            if (j < (long long)TOT_)
                write_row(out, anchors, gt, cls, b, j, a * G_ + g, 0.0f);
            ++k;
        }
    }
}

// Zero only the rows not covered by pos/neg (normally none).
__global__ void tail_fill_kernel(const unsigned* __restrict__ counters,
                                 float* __restrict__ out)
{
    const int b = blockIdx.y;
    const unsigned j = blockIdx.x * BDIM + threadIdx.x;
    const unsigned long long end =
        (unsigned long long)counters[b * 8 + 1] + counters[b * 8 + 2];
    if (j < (unsigned)TOT_ && (unsigned long long)j >= end) {
        const size_t NT = (size_t)B_ * TOT_;
        const size_t r = (size_t)b * TOT_ + j;
        ((float4*)out)[r] = make_float4(0.f, 0.f, 0.f, 0.f);
        out[NT * 4 + r] = 0.f;
        out[NT * 5 + r] = 0.f;
        ((float4*)(out + NT * 6))[r] = make_float4(0.f, 0.f, 0.f, 0.f);
    }
}

__global__ void zero_counters_kernel(unsigned* __restrict__ counters)
{
    if (threadIdx.x < (unsigned)(B_ * 8)) counters[threadIdx.x] = 0u;
}

// ---------------------------------------------------------------------------
extern "C" void kernel_launch(void* const* d_in, const int* in_sizes, int n_in,
                              void* d_out, int out_size, void* d_ws, size_t ws_size,
                              hipStream_t stream)
{
    const float* anchors = (const float*)d_in[0];   // (B, A, 4)
    const float* gt      = (const float*)d_in[1];   // (B, G, 4) unscaled
    const int*   cls     = (const int*)d_in[2];     // (B, G)
    float* out = (float*)d_out;
    char*  ws  = (char*)d_ws;

    const size_t offNegcnt  = 0;
    const size_t offNegmask = offNegcnt  + 4ull * B_ * A_;
    const size_t offNegpre  = offNegmask + 4ull * B_ * A_;
    const size_t offBlock   = offNegpre  + 4ull * B_ * A_;
    const size_t offPosiou  = offBlock   + 4ull * B_ * NBLK_;
    const size_t offPosidx  = offPosiou  + 4ull * B_ * POS_CAP_;
    const size_t offPossort = offPosidx  + 4ull * B_ * POS_CAP_;
    const size_t offCnt     = offPossort + 4ull * B_ * POS_CAP_;

    unsigned* negcnt   = (unsigned*)(ws + offNegcnt);
    unsigned* negmask  = (unsigned*)(ws + offNegmask);
    unsigned* negpre   = (unsigned*)(ws + offNegpre);
    unsigned* blocksum = (unsigned*)(ws + offBlock);
    float*    posiou   = (float*)   (ws + offPosiou);
    unsigned* posidx   = (unsigned*)(ws + offPosidx);
    unsigned* possort  = (unsigned*)(ws + offPossort);
    unsigned* counters = (unsigned*)(ws + offCnt);

    zero_counters_kernel<<<1, 32, 0, stream>>>(counters);
    iou_mask_kernel<<<dim3(GRIDX_, B_), BDIM, 0, stream>>>(
        anchors, gt, negcnt, negmask, posiou, posidx, counters);
    scan_block_kernel<<<dim3(NBLK_, B_), BDIM, 0, stream>>>(negcnt, negpre, blocksum);
    scan_top_kernel<<<B_, NBLK_, 0, stream>>>(blocksum, counters);
    sort_pos_kernel<<<B_, BDIM, 0, stream>>>(posiou, posidx, possort, counters);
    tail_fill_kernel<<<dim3(NBLK_, B_), BDIM, 0, stream>>>(counters, out);
    write_pos_kernel<<<dim3((POS_CAP_ + BDIM - 1) / BDIM, B_), BDIM, 0, stream>>>(
        anchors, gt, cls, possort, counters, out);
    write_neg_kernel<<<dim3(NBLK_, B_), BDIM, 0, stream>>>(
        anchors, gt, cls, negpre, blocksum, negmask, counters, out);
}